// TokenEmbedding_26980984554065
// MI455X (gfx1250) — compile-verified
//
#include <hip/hip_runtime.h>

typedef __attribute__((ext_vector_type(2))) float v2f;
typedef __attribute__((ext_vector_type(8))) float v8f;

#define TAO    3
#define NLAG   6          // M+1
#define CIN    7
#define DMODEL 512
#define KCH    73         // KERNELS = 512/7
#define LN     8192
#define BB     32
#define KRED   18         // NLAG*3 reduction length
#define NSTEP  5          // ceil(18/4) WMMA K-steps
#define NKT    5          // ceil(74/16) k tiles
#define TPW    2          // t-tiles (of 16) per wave
#define CHUNKS (LN / 16 / TPW)   // 256 chunks per (b,c)
#define NWAVES (BB * CIN * CHUNKS)      // 57344
#define BLKDIM 256
#define NBLOCK (NWAVES * 32 / BLKDIM)   // 7168

__global__ __launch_bounds__(BLKDIM) void lagconv_wmma_f32(
    const float* __restrict__ x,        // [B, L, CIN]
    const float* __restrict__ conv_w,   // [73, 6, 3]
    const float* __restrict__ conv_b,   // [73]
    const float* __restrict__ left_w,   // [1, 6, 3]
    const float* __restrict__ left_b,   // [1]
    float* __restrict__ out)            // [B, L, 512]
{
  const int lane  = threadIdx.x & 31;
  const int wid   = (int)((blockIdx.x * blockDim.x + threadIdx.x) >> 5);
  const int bc    = wid / CHUNKS;          // 0 .. 223
  const int chunk = wid % CHUNKS;          // 0 .. 255
  const int b     = bc / CIN;
  const int c     = bc % CIN;
  const int kmax  = (c == CIN - 1) ? (KCH + 1) : KCH;  // fold rem channel into c==6

  const int col  = lane & 15;              // N index (output channel within tile)
  const int half = lane >> 4;              // hi/lo half of wave
  const int trow = half * 8;               // C/D row offset for this half

  // ---- Build B tiles (K x N per WMMA step) and per-column bias, kept in VGPRs ----
  v2f   Breg[NKT][NSTEP];
  float bias[NKT];
#pragma unroll
  for (int kt = 0; kt < NKT; ++kt) {
    const int k = kt * 16 + col;
    bias[kt] = (k < KCH) ? conv_b[k]
             : ((k == KCH && c == CIN - 1) ? left_b[0] : 0.0f);
#pragma unroll
    for (int kk = 0; kk < NSTEP; ++kk) {
#pragma unroll
      for (int r = 0; r < 2; ++r) {
        const int mj = kk * 4 + r + 2 * half;     // reduction index = m*3 + j
        float wv = 0.0f;
        if (mj < KRED && k < kmax) {
          const int m = mj / 3, j = mj % 3;
          wv = (k < KCH) ? conv_w[(k * NLAG + m) * 3 + j] : left_w[m * 3 + j];
        }
        Breg[kt][kk][r] = wv;
      }
    }
  }

  const float* xb = x + (size_t)b * LN * CIN + c;   // x[b, :, c], stride CIN

  for (int tt = 0; tt < TPW; ++tt) {
    const int t0 = (chunk * TPW + tt) * 16;
    const int tA = t0 + col;                        // A-matrix row for this lane

    // ---- Build A tile: A[t, mj] = feat(m, (t+j-1) mod n) ----
    v2f Areg[NSTEP];
#pragma unroll
    for (int kk = 0; kk < NSTEP; ++kk) {
#pragma unroll
      for (int r = 0; r < 2; ++r) {
        const int mj = kk * 4 + r + 2 * half;
        float av = 0.0f;
        if (mj < KRED) {
          const int m = mj / 3, j = mj % 3;
          int s = tA + j - 1;
          if (s < 0)   s += LN;
          if (s >= LN) s -= LN;
          if (s >= (NLAG - 1) * TAO)                // s >= 15: valid lag window
            av = xb[(size_t)(s - m * TAO) * CIN];
        }
        Areg[kk][r] = av;
      }
    }

    // ---- 5 k-tiles x 5 K-steps of V_WMMA_F32_16X16X4_F32 ----
#pragma unroll
    for (int kt = 0; kt < NKT; ++kt) {
      v8f acc;
#pragma unroll
      for (int r = 0; r < 8; ++r) acc[r] = bias[kt];  // C = bias broadcast per column
#pragma unroll
      for (int kk = 0; kk < NSTEP; ++kk) {
        acc = __builtin_amdgcn_wmma_f32_16x16x4_f32(
            /*neg_a=*/false, Areg[kk],
            /*neg_b=*/false, Breg[kt][kk],
            /*c_mod=*/(short)0, acc,
            /*reuse_a=*/false, /*reuse_b=*/false);
      }

      const int k = kt * 16 + col;
      if (k < kmax) {
        float* op = out + (size_t)b * LN * DMODEL + (size_t)c * KCH + k;
#pragma unroll
        for (int r = 0; r < 8; ++r) {
          // Output is 512 MB, written once, never re-read: stream past L2
          // (TH=NT) so x (7.3 MB, ~18x reuse) and weights stay cache-resident.
          __builtin_nontemporal_store(acc[r], &op[(size_t)(t0 + trow + r) * DMODEL]);
        }
      }
    }
  }
}

extern "C" void kernel_launch(void* const* d_in, const int* in_sizes, int n_in,
                              void* d_out, int out_size, void* d_ws, size_t ws_size,
                              hipStream_t stream) {
  (void)in_sizes; (void)n_in; (void)d_ws; (void)ws_size; (void)out_size;
  const float* x      = (const float*)d_in[0];
  const float* conv_w = (const float*)d_in[1];
  const float* conv_b = (const float*)d_in[2];
  const float* left_w = (const float*)d_in[3];
  const float* left_b = (const float*)d_in[4];
  float* out = (float*)d_out;

  lagconv_wmma_f32<<<NBLOCK, BLKDIM, 0, stream>>>(x, conv_w, conv_b,
                                                  left_w, left_b, out);
}